// NCCLoss_79740362817967
// MI455X (gfx1250) — compile-verified
//
#include <hip/hip_runtime.h>

// ---------------------------------------------------------------------------
// Fused 3D NCC loss for MI455X (gfx1250, wave32).
//   - one pass over I/J (memory-bound: ~146MB @ 23.3TB/s ≈ 6.3us)
//   - W box-sum: VALU sliding window
//   - H box-sum: V_WMMA_F32_16X16X4_F32 banded-ones GEMM (matrix pipe, f32-exact)
//   - D box-sum: running window with 7-deep LDS ring
//   - cc via v_rcp_f32 (no IEEE div sequences in the hot loop)
//   - block reduction + single atomicAdd per block
// ---------------------------------------------------------------------------

typedef __attribute__((ext_vector_type(2))) float v2f;
typedef __attribute__((ext_vector_type(8))) float v8f;

#define B_   2
#define D_   128
#define H_   192
#define W_   192
#define TILE 16
#define HALO 22      // TILE + 6
#define KPAD 24      // HALO padded to multiple of 4 (WMMA K-step)
#define NCH  5
#define WIN  7
#define NTOT (B_ * D_ * H_ * W_)   // 9437184

__global__ void ncc_init(float* out) { out[0] = 0.0f; }

__device__ __forceinline__ float band(int m, int k) {
    // out row m accumulates input rows m..m+6
    return (k >= m && k <= m + 6) ? 1.0f : 0.0f;
}

__global__ __launch_bounds__(256)
void ncc_fused(const float* __restrict__ I, const float* __restrict__ J,
               float* __restrict__ out)
{
    __shared__ float lds_ch  [NCH * HALO * HALO];   // 5 channel slices, 22x22
    __shared__ float lds_wsum[NCH * KPAD * TILE];   // W-summed rows (rows 22..23 = 0)
    __shared__ float lds_ring[WIN * NCH * 256];     // 7-slice D ring of HW-summed tiles
    __shared__ float lds_fin [NCH * 256];           // current D-window sums, all 5 ch
    __shared__ float lds_red [256];

    const int tid  = threadIdx.x;
    const int lane = tid & 31;
    const int wave = tid >> 5;
    const int half = lane >> 4;      // 0: lanes 0-15, 1: lanes 16-31
    const int m16  = lane & 15;

    const int b  = blockIdx.z;
    const int h0 = blockIdx.y * TILE;
    const int w0 = blockIdx.x * TILE;

    const long long planeHW = (long long)H_ * W_;
    const float* Ib = I + (long long)b * D_ * planeHW;
    const float* Jb = J + (long long)b * D_ * planeHW;

    // zero wsum once: rows 22..23 stay zero forever (WMMA K padding)
    for (int i = tid; i < NCH * KPAD * TILE; i += 256) lds_wsum[i] = 0.0f;

    v8f runsum = {};
    float acc_total = 0.0f;

    // ---- compute one z-slice: load+channels -> W-sum -> H-sum (WMMA) ----
    auto compute_slice = [&](int s) -> v8f {
        // stage 1: global -> 5 channel LDS slices (halo'd, zero padded)
        for (int idx = tid; idx < HALO * HALO; idx += 256) {
            int r = idx / HALO, c = idx % HALO;
            int gh = h0 - 3 + r, gw = w0 - 3 + c;
            float vi = 0.0f, vj = 0.0f;
            if (gh >= 0 && gh < H_ && gw >= 0 && gw < W_) {
                long long off = (long long)s * planeHW + (long long)gh * W_ + gw;
                vi = Ib[off];
                vj = Jb[off];
                if (s + 1 < D_) {                  // prefetch next slice row
                    __builtin_prefetch(Ib + off + planeHW, 0, 1);
                    __builtin_prefetch(Jb + off + planeHW, 0, 1);
                }
            }
            lds_ch[0 * (HALO * HALO) + idx] = vi;
            lds_ch[1 * (HALO * HALO) + idx] = vj;
            lds_ch[2 * (HALO * HALO) + idx] = vi * vi;
            lds_ch[3 * (HALO * HALO) + idx] = vj * vj;
            lds_ch[4 * (HALO * HALO) + idx] = vi * vj;
        }
        __syncthreads();  // B1

        // stage 2: W-direction sliding 7-sum (110 worker lanes)
        if (tid < NCH * HALO) {
            int ch = tid / HALO, row = tid % HALO;
            const float* src = &lds_ch[ch * (HALO * HALO) + row * HALO];
            float* dst = &lds_wsum[(ch * KPAD + row) * TILE];
            float s7 = src[0] + src[1] + src[2] + src[3] + src[4] + src[5] + src[6];
            dst[0] = s7;
            for (int w = 1; w < TILE; ++w) {
                s7 += src[w + 6] - src[w - 1];
                dst[w] = s7;
            }
        }
        __syncthreads();  // B2

        // stage 3: H-direction 7-sum as banded-ones GEMM on the WMMA pipe.
        //   D(16x16) = Band(16x24) x Wsum(24x16), six K=4 steps, f32-exact.
        v8f acc = {};
        if (wave < NCH) {
            const float* bs = &lds_wsum[wave * KPAD * TILE];
#pragma unroll
            for (int kk = 0; kk < 6; ++kk) {
                int k0 = kk * 4 + 2 * half;        // A/B lane K layout (ISA 7.12.2)
                v2f a, bm;
                a.x  = band(m16, k0);
                a.y  = band(m16, k0 + 1);
                bm.x = bs[(k0)     * TILE + m16];
                bm.y = bs[(k0 + 1) * TILE + m16];
                acc = __builtin_amdgcn_wmma_f32_16x16x4_f32(
                        false, a, false, bm, (short)0, acc, false, false);
            }
        }
        return acc;
    };

    // ---- prologue: slices 0..2 into the running D-window ----
    for (int s = 0; s < 3; ++s) {
        v8f t = compute_slice(s);
        if (wave < NCH) {
            float* rg = &lds_ring[((s % WIN) * NCH + wave) * 256];
#pragma unroll
            for (int r = 0; r < 8; ++r) rg[lane * 8 + r] = t[r];
            runsum += t;
        }
    }

    // ---- main D loop ----
    for (int d = 0; d < D_; ++d) {
        int s = d + 3;
        if (s < D_) {                              // block-uniform branch
            v8f t = compute_slice(s);
            if (wave < NCH) {
                float* rg = &lds_ring[((s % WIN) * NCH + wave) * 256];
#pragma unroll
                for (int r = 0; r < 8; ++r) rg[lane * 8 + r] = t[r];
                runsum += t;
            }
        }

        // publish the 5 channel window-sums (C/D layout: M = r + 8*half, N = m16)
        if (wave < NCH) {
            float* fin = &lds_fin[wave * 256];
#pragma unroll
            for (int r = 0; r < 8; ++r) fin[(r + 8 * half) * TILE + m16] = runsum[r];
        }
        __syncthreads();  // B3

        // cc for this thread's (ty,tx) element at depth d
        {
            int e = tid;                           // ty*16+tx == tid for 256 threads
            float Is = lds_fin[0 * 256 + e];
            float Js = lds_fin[1 * 256 + e];
            float I2 = lds_fin[2 * 256 + e];
            float J2 = lds_fin[3 * 256 + e];
            float IJ = lds_fin[4 * 256 + e];
            const float ws     = 343.0f;           // 7^3
            const float inv_ws = 1.0f / 343.0f;    // exact-constant reciprocal
            float uI = Is * inv_ws, uJ = Js * inv_ws;
            float cross = IJ - uJ * Is - uI * Js + uI * uJ * ws;
            float Iv = I2 - 2.0f * uI * Is + uI * uI * ws;
            float Jv = J2 - 2.0f * uJ * Js + uJ * uJ * ws;
            // v_rcp_f32 (<=1 ulp) instead of IEEE div sequence
            float rden = __builtin_amdgcn_rcpf(Iv * Jv + 1e-5f);
            acc_total += cross * cross * rden;
        }
        __syncthreads();  // B4 (lds_fin reuse)

        // retire slice leaving the window of d+1
        int old = d - 3;
        if (old >= 0 && wave < NCH) {
            const float* rg = &lds_ring[((old % WIN) * NCH + wave) * 256];
            v8f t;
#pragma unroll
            for (int r = 0; r < 8; ++r) t[r] = rg[lane * 8 + r];
            runsum -= t;
        }
    }

    // ---- block reduction + global accumulate of -mean ----
    lds_red[tid] = acc_total;
    __syncthreads();
    for (int off = 128; off > 0; off >>= 1) {
        if (tid < off) lds_red[tid] += lds_red[tid + off];
        __syncthreads();
    }
    if (tid == 0)
        atomicAdd(out, -lds_red[0] * (1.0f / (float)NTOT));
}

extern "C" void kernel_launch(void* const* d_in, const int* in_sizes, int n_in,
                              void* d_out, int out_size, void* d_ws, size_t ws_size,
                              hipStream_t stream) {
    (void)in_sizes; (void)n_in; (void)out_size; (void)d_ws; (void)ws_size;
    const float* I = (const float*)d_in[0];
    const float* J = (const float*)d_in[1];
    float* out = (float*)d_out;

    ncc_init<<<dim3(1), dim3(1), 0, stream>>>(out);
    dim3 grid(W_ / TILE, H_ / TILE, B_);   // 12 x 12 x 2 = 288 blocks
    ncc_fused<<<grid, dim3(256), 0, stream>>>(I, J, out);
}